// model_GUNet_27736898798371
// MI455X (gfx1250) — compile-verified
//
#include <hip/hip_runtime.h>
#include <hip/hip_bf16.h>

typedef float v2f __attribute__((ext_vector_type(2)));
typedef float v8f __attribute__((ext_vector_type(8)));

#define CDIV(a, b) (((a) + (b) - 1) / (b))

// ---------------------------------------------------------------------------
// SpMM: Y[m,c] = sum_k pow(A[R(m), R(k)], 2^NSQ) * Z[k,c] (+bias[c]) (+addend)
// R = idx (composed top-k gather, HASIDX=true) or identity.
// V_WMMA_F32_16X16X4_F32; one wave per 16-row tile; CT 16-wide column tiles.
// All nK are multiples of 4; M tail via clamped loads + guarded stores so
// EXEC is all-ones through the WMMAs. HASIDX/NSQ are compile-time so the
// K loop is branch-free straight-line code.
// ---------------------------------------------------------------------------
template <int CT, bool HASIDX, int NSQ>
__global__ __launch_bounds__(256) void spmm_wmma_kernel(
    const float* __restrict__ A, int lda, const int* __restrict__ idx,
    const float* __restrict__ Z, int ldz, int nM, int nK,
    float* __restrict__ Y, int ldy, const float* __restrict__ bias,
    const float* __restrict__ addend, int ldadd) {
  const int lane = threadIdx.x & 31;
  const int wave = threadIdx.x >> 5;
  const int tile = blockIdx.x * (blockDim.x >> 5) + wave;
  const int m0 = tile * 16;
  if (m0 >= nM) return;  // wave-uniform
  const int lm = lane & 15;
  const int kh = lane >> 4;

  int r = m0 + lm;
  if (r >= nM) r = nM - 1;  // clamp (no divergence); stores guarded below
  const long grow = HASIDX ? (long)idx[r] : (long)r;
  const float* __restrict__ rowPtr = A + grow * (long)lda;

  v8f acc[CT];
#pragma unroll
  for (int t = 0; t < CT; ++t) {
    v8f z = {};
    acc[t] = z;
  }

  const float* __restrict__ aPtr = rowPtr + 2 * kh;   // identity path
  const int* __restrict__ cPtr = idx + 2 * kh;        // gather path
  const float* __restrict__ zPtr = Z + (size_t)(2 * kh) * ldz + lm;

#pragma unroll 2
  for (int k = 0; k < nK; k += 4) {
    // A fragment: lane holds A[M=lm, K = k + 2*kh + {0,1}] (32-bit 16x4 layout)
    v2f a;
    if (HASIDX) {
      int2 cc = *(const int2*)(cPtr + k);
      a.x = rowPtr[cc.x];
      a.y = rowPtr[cc.y];
    } else {
      a = *(const v2f*)(aPtr + k);
    }
#pragma unroll
    for (int s = 0; s < NSQ; ++s) a *= a;  // A^(2^NSQ) elementwise

    const float* zb = zPtr + (size_t)k * ldz;
#pragma unroll
    for (int t = 0; t < CT; ++t) {
      // B fragment (4x16): vgpr r holds Z[K = k + r + 2*kh, N = lm + 16*t]
      v2f b;
      b.x = zb[t * 16];
      b.y = zb[(size_t)ldz + t * 16];
      acc[t] = __builtin_amdgcn_wmma_f32_16x16x4_f32(false, a, false, b,
                                                     (short)0, acc[t], false,
                                                     false);
    }
  }

  // C/D layout: vgpr rr -> (M = m0 + rr + 8*kh, N = lm + 16*t)
#pragma unroll
  for (int t = 0; t < CT; ++t) {
#pragma unroll
    for (int rr = 0; rr < 8; ++rr) {
      int m = m0 + rr + 8 * kh;
      if (m < nM) {
        int col = t * 16 + lm;
        float v = acc[t][rr];
        if (bias) v += bias[col];
        if (addend) v += addend[(size_t)m * ldadd + col];
        Y[(size_t)m * ldy + col] = v;
      }
    }
  }
}

// ---------------------------------------------------------------------------
// Small dense GEMM: Y = X @ W (+bias), one thread per output element.
// ---------------------------------------------------------------------------
__global__ void gemm_small_kernel(const float* __restrict__ X, int ldx,
                                  const float* __restrict__ W, int ldw,
                                  const float* __restrict__ bias,
                                  float* __restrict__ Y, int ldy, int n,
                                  int cin, int cout) {
  int i = blockIdx.x * blockDim.x + threadIdx.x;
  if (i >= n * cout) return;
  int row = i / cout, c = i % cout;
  float s = bias ? bias[c] : 0.0f;
  const float* xr = X + (size_t)row * ldx;
  for (int k = 0; k < cin; ++k) s += xr[k] * W[(size_t)k * ldw + c];
  Y[(size_t)row * ldy + c] = s;
}

// ---------------------------------------------------------------------------
// Pool: scores = sigmoid(Y @ wp + bp); exact top-k via radix select on float
// bit patterns (scores > 0 so bits are order-isomorphic); ties broken by
// smallest index (matches jax.lax.top_k). Emits selected indices sorted
// ascending (a consistent re-permutation of the level => mathematically
// equivalent), composed global indices, and paired scores. One block, LDS-only,
// scan-based (no atomics) => deterministic.
// ---------------------------------------------------------------------------
__device__ int block_excl_scan_1024(int v, int* s) {
  const int t = threadIdx.x;
  s[t] = v;
  __syncthreads();
  for (int off = 1; off < 1024; off <<= 1) {
    int x = s[t];
    int y = (t >= off) ? s[t - off] : 0;
    __syncthreads();
    s[t] = x + y;
    __syncthreads();
  }
  int incl = s[t];
  __syncthreads();
  return incl - v;
}

__global__ __launch_bounds__(1024) void pool_kernel(
    const float* __restrict__ Y, int ldy, const float* __restrict__ wp,
    const float* __restrict__ bp, const int* __restrict__ prevGlob, int n,
    int k, int* __restrict__ selIdx, int* __restrict__ selGlob,
    float* __restrict__ vs) {
  __shared__ float s_sc[8192];
  __shared__ int s_red[1024];
  const int t = threadIdx.x;
  const float b0 = bp[0];
  for (int r = t; r < n; r += 1024) {
    float d = b0;
    const float* yr = Y + (size_t)r * ldy;
#pragma unroll
    for (int c = 0; c < 48; ++c) d += yr[c] * wp[c];
    s_sc[r] = 1.0f / (1.0f + __expf(-d));
  }
  __syncthreads();

  // radix select the k-th largest bit pattern
  unsigned pref = 0, mask = 0;
  int need = k;
  for (int b = 31; b >= 0; --b) {
    unsigned m2 = mask | (1u << b);
    unsigned p2 = pref | (1u << b);
    int cnt = 0;
    for (int r = t; r < n; r += 1024) {
      unsigned u = __float_as_uint(s_sc[r]);
      cnt += ((u & m2) == p2) ? 1 : 0;
    }
    s_red[t] = cnt;
    __syncthreads();
    for (int off = 512; off > 0; off >>= 1) {
      if (t < off) s_red[t] += s_red[t + off];
      __syncthreads();
    }
    int tot = s_red[0];
    __syncthreads();
    if (tot >= need)
      pref = p2;
    else
      need -= tot;
    mask = m2;
  }
  // select: all > pref, plus first `need` (by index) equal to pref
  const int chunk = (n + 1023) / 1024;
  const int lo = t * chunk;
  const int hi = (lo + chunk < n) ? (lo + chunk) : n;
  int cEq = 0;
  for (int r = lo; r < hi; ++r) cEq += (__float_as_uint(s_sc[r]) == pref);
  int eqBase = block_excl_scan_1024(cEq, s_red);
  int selCnt = 0;
  {
    int eqr = eqBase;
    for (int r = lo; r < hi; ++r) {
      unsigned u = __float_as_uint(s_sc[r]);
      if (u > pref)
        selCnt++;
      else if (u == pref) {
        if (eqr < need) selCnt++;
        eqr++;
      }
    }
  }
  int pos = block_excl_scan_1024(selCnt, s_red);
  {
    int eqr = eqBase;
    for (int r = lo; r < hi; ++r) {
      unsigned u = __float_as_uint(s_sc[r]);
      bool sel = false;
      if (u > pref)
        sel = true;
      else if (u == pref) {
        sel = (eqr < need);
        eqr++;
      }
      if (sel) {
        selIdx[pos] = r;
        vs[pos] = s_sc[r];
        selGlob[pos] = prevGlob ? prevGlob[r] : r;
        pos++;
      }
    }
  }
}

// x_next[j,:] = Y[sel[j],:] * vs[j]
__global__ void gather_scale_kernel(const float* __restrict__ Y, int ldy,
                                    const int* __restrict__ sel,
                                    const float* __restrict__ vs, int k,
                                    float* __restrict__ out) {
  int i = blockIdx.x * blockDim.x + threadIdx.x;
  if (i >= k * 48) return;
  int j = i / 48, c = i % 48;
  out[(size_t)j * 48 + c] = Y[(size_t)sel[j] * ldy + c] * vs[j];
}

__global__ void zero_fill_kernel(float* __restrict__ p, size_t n) {
  size_t i = (size_t)blockIdx.x * blockDim.x + threadIdx.x;
  if (i < n) p[i] = 0.0f;
}

// Zpad[sel[j],:] = Z[j,:]
__global__ void scatter_rows_kernel(const float* __restrict__ Z,
                                    const int* __restrict__ sel, int k,
                                    float* __restrict__ Zpad) {
  int i = blockIdx.x * blockDim.x + threadIdx.x;
  if (i >= k * 48) return;
  int j = i / 48, c = i % 48;
  Zpad[(size_t)sel[j] * 48 + c] = Z[(size_t)j * 48 + c];
}

// ---------------------------------------------------------------------------
extern "C" void kernel_launch(void* const* d_in, const int* in_sizes, int n_in,
                              void* d_out, int out_size, void* d_ws,
                              size_t ws_size, hipStream_t stream) {
  (void)in_sizes; (void)n_in; (void)out_size; (void)ws_size;
  const float* A       = (const float*)d_in[0];
  const float* x       = (const float*)d_in[1];
  const float* W_start = (const float*)d_in[2];
  const float* b_start = (const float*)d_in[3];
  const float* W_down  = (const float*)d_in[4];
  const float* b_down  = (const float*)d_in[5];
  const float* W_pool  = (const float*)d_in[6];
  const float* b_pool  = (const float*)d_in[7];
  const float* W_bot   = (const float*)d_in[8];
  const float* b_bot   = (const float*)d_in[9];
  const float* W_up    = (const float*)d_in[10];
  const float* b_up    = (const float*)d_in[11];
  const float* W_end   = (const float*)d_in[12];
  const float* b_end   = (const float*)d_in[13];
  float* out = (float*)d_out;

  // static level sizes: n -> floor(KS[i]*n)
  const int nl[5] = {8192, 7372, 5160, 3096, 1548};

  // workspace carve-up (all counts even -> int2 loads stay 8B aligned)
  float* w = (float*)d_ws;
  size_t o = 0;
  float* Zb   = w + o; o += (size_t)8192 * 48;
  float* Zpad = w + o; o += (size_t)8192 * 48;
  float* xA   = w + o; o += (size_t)8192 * 48;
  float* xB   = w + o; o += (size_t)8192 * 48;
  float* xcat = w + o; o += (size_t)8192 * 96;  // [up result | org_x]
  float* Tend = w + o; o += (size_t)8192 * 96;
  float* down[4];
  for (int i = 0; i < 4; ++i) { down[i] = w + o; o += (size_t)nl[i] * 48; }
  float* vsA[4];
  for (int i = 0; i < 4; ++i) { vsA[i] = w + o; o += (size_t)nl[i + 1]; }
  int* ip = (int*)(w + o);
  int* selIdx[4];
  int* selGlob[4];
  for (int i = 0; i < 4; ++i) { selIdx[i] = ip; ip += nl[i + 1]; }
  for (int i = 0; i < 4; ++i) { selGlob[i] = ip; ip += nl[i + 1]; }

  // CT=3 SpMM dispatch over the 5 live (HASIDX, NSQ) configs
  auto spmm3 = [&](const int* idx, int nsq, const float* Z, int ldz, int n,
                   float* Y, int ldy, const float* bias, const float* addend) {
    int blocks = CDIV(CDIV(n, 16), 8);
    if (!idx) {
      spmm_wmma_kernel<3, false, 0><<<blocks, 256, 0, stream>>>(
          A, 8192, nullptr, Z, ldz, n, n, Y, ldy, bias, addend, 48);
      return;
    }
    switch (nsq) {
      case 1:
        spmm_wmma_kernel<3, true, 1><<<blocks, 256, 0, stream>>>(
            A, 8192, idx, Z, ldz, n, n, Y, ldy, bias, addend, 48);
        break;
      case 2:
        spmm_wmma_kernel<3, true, 2><<<blocks, 256, 0, stream>>>(
            A, 8192, idx, Z, ldz, n, n, Y, ldy, bias, addend, 48);
        break;
      case 3:
        spmm_wmma_kernel<3, true, 3><<<blocks, 256, 0, stream>>>(
            A, 8192, idx, Z, ldz, n, n, Y, ldy, bias, addend, 48);
        break;
      default:
        spmm_wmma_kernel<3, true, 4><<<blocks, 256, 0, stream>>>(
            A, 8192, idx, Z, ldz, n, n, Y, ldy, bias, addend, 48);
        break;
    }
  };

  // ---- start: org = A @ (x @ W_start) + b_start  -> xcat[:,48:96]
  gemm_small_kernel<<<CDIV(8192 * 48, 256), 256, 0, stream>>>(
      x, 128, W_start, 48, nullptr, Zb, 48, 8192, 128, 48);
  spmm3(nullptr, 0, Zb, 48, 8192, xcat + 48, 96, b_start, nullptr);

  // ---- down path
  const float* xin = xcat + 48;
  int ldxin = 96;
  for (int i = 0; i < 4; ++i) {
    int n = nl[i], k = nl[i + 1];
    gemm_small_kernel<<<CDIV(n * 48, 256), 256, 0, stream>>>(
        xin, ldxin, W_down + (size_t)i * 48 * 48, 48, nullptr, Zb, 48, n, 48,
        48);
    spmm3(i ? selGlob[i - 1] : nullptr, i, Zb, 48, n, down[i], 48,
          b_down + (size_t)i * 48, nullptr);
    pool_kernel<<<1, 1024, 0, stream>>>(down[i], 48, W_pool + (size_t)i * 48,
                                        b_pool + i,
                                        i ? selGlob[i - 1] : nullptr, n, k,
                                        selIdx[i], selGlob[i], vsA[i]);
    gather_scale_kernel<<<CDIV(k * 48, 256), 256, 0, stream>>>(
        down[i], 48, selIdx[i], vsA[i], k, xA);
    xin = xA;
    ldxin = 48;
  }

  // ---- bottom: A_4 = A[I4,I4]^16
  gemm_small_kernel<<<CDIV(nl[4] * 48, 256), 256, 0, stream>>>(
      xA, 48, W_bot, 48, nullptr, Zb, 48, nl[4], 48, 48);
  spmm3(selGlob[3], 4, Zb, 48, nl[4], xB, 48, b_bot, nullptr);
  const float* xcur = xB;

  // ---- up path
  for (int i = 0; i < 4; ++i) {
    int u = 3 - i;
    int n = nl[u], k = nl[u + 1];
    gemm_small_kernel<<<CDIV(k * 48, 256), 256, 0, stream>>>(
        xcur, 48, W_up + (size_t)i * 48 * 48, 48, nullptr, Zb, 48, k, 48, 48);
    zero_fill_kernel<<<CDIV((size_t)n * 48, 256), 256, 0, stream>>>(
        Zpad, (size_t)n * 48);
    scatter_rows_kernel<<<CDIV(k * 48, 256), 256, 0, stream>>>(Zb, selIdx[u],
                                                               k, Zpad);
    float* yout;
    int ldy;
    if (u == 0) {
      yout = xcat;  // left half of concat buffer
      ldy = 96;
    } else {
      yout = (xcur == xA) ? xB : xA;
      ldy = 48;
    }
    spmm3(u ? selGlob[u - 1] : nullptr, u, Zpad, 48, n, yout, ldy,
          b_up + (size_t)i * 48, down[u]);
    xcur = yout;
  }

  // ---- end: out = (A @ [x_up | org]) @ W_end + b_end
  {
    int blocks = CDIV(CDIV(8192, 16), 8);
    spmm_wmma_kernel<6, false, 0><<<blocks, 256, 0, stream>>>(
        A, 8192, nullptr, xcat, 96, 8192, 8192, Tend, 96, nullptr, nullptr,
        96);
  }
  gemm_small_kernel<<<CDIV(8192 * 128, 256), 256, 0, stream>>>(
      Tend, 96, W_end, 128, b_end, out, 128, 8192, 96, 128);
}